// DAHH_11639361372555
// MI455X (gfx1250) — compile-verified
//
#include <hip/hip_runtime.h>
#include <cmath>

#define B_   4
#define C_   768
#define L_   2048
#define OUT_ 159
#define NT_  10                      // ceil(OUT/16) N-tiles
#define MT_  ((B_ * L_) / 16)        // 512 M-tiles over all batches
#define BN_EPS 1e-5f

typedef __attribute__((ext_vector_type(2))) float v2f;
typedef __attribute__((ext_vector_type(8))) float v8f;
typedef int v4i_ __attribute__((vector_size(4 * sizeof(int))));
typedef __attribute__((address_space(1))) v4i_* gvec_ptr;   // global b128
typedef __attribute__((address_space(3))) v4i_* lvec_ptr;   // LDS b128

#if defined(__has_builtin)
#if __has_builtin(__builtin_amdgcn_global_load_async_to_lds_b128)
#define HAVE_ASYNC_COPY 1
#endif
#if __has_builtin(__builtin_amdgcn_s_wait_asynccnt)
#define HAVE_ASYNC_WAIT 1
#endif
#endif

static __device__ __forceinline__ v8f wmma_f32_16x16x4(v2f a, v2f b, v8f c) {
  // D = A(16x4) * B(4x16) + C(16x16), fp32, wave32
  return __builtin_amdgcn_wmma_f32_16x16x4_f32(false, a, false, b, (short)0, c,
                                               false, false);
}

// Stage `n4` float4s from global `src` into LDS `dst` cooperatively.
// Uses the gfx1250 async global->LDS path (ASYNCcnt) when available.
static __device__ __forceinline__ void stage_to_lds(const float* __restrict__ src,
                                                    float* __restrict__ dst,
                                                    int n4, int tid, int nthr) {
#if HAVE_ASYNC_COPY
  for (int i = tid; i < n4; i += nthr) {
    __builtin_amdgcn_global_load_async_to_lds_b128(
        (gvec_ptr)(unsigned long long)(src + 4 * i),
        (lvec_ptr)(dst + 4 * i), 0, 0);
  }
#if HAVE_ASYNC_WAIT
  __builtin_amdgcn_s_wait_asynccnt(0);
#else
  asm volatile("s_wait_asynccnt 0" ::: "memory");
#endif
#else
  const float4* s4 = (const float4*)src;
  float4* d4 = (float4*)dst;
  for (int i = tid; i < n4; i += nthr) d4[i] = s4[i];
#endif
}

// ---------------------------------------------------------------- sq per row
__global__ void sq_kernel(const float* __restrict__ x, float* __restrict__ sq) {
  int idx = blockIdx.x * blockDim.x + threadIdx.x;   // row in (B*L, C)
  if (idx >= B_ * L_) return;
  const float4* row = (const float4*)(x + (size_t)idx * C_);
  float s = 0.f;
  for (int c = 0; c < C_ / 4; ++c) {
    float4 v = row[c];
    s += v.x * v.x + v.y * v.y + v.z * v.z + v.w * v.w;
  }
  sq[idx] = s;
}

// ------------------------------------------------- xt = x(B*L,C) @ theta(C,OUT)
// Block = 8 waves sharing one zero-padded, transposed theta N-panel in LDS.
__global__ void xt_gemm_kernel(const float* __restrict__ x,
                               const float* __restrict__ theta,
                               float* __restrict__ xt) {
  __shared__ float sT[16 * C_];                // theta^T panel [n][k], 48 KB

  int nt = blockIdx.y;                         // N-tile 0..9
  // stage + transpose + zero-pad the theta panel once per block
  for (int i = threadIdx.x; i < 16 * C_; i += blockDim.x) {
    int n = i / C_;
    int k = i - n * C_;
    int gn = nt * 16 + n;
    sT[i] = (gn < OUT_) ? theta[(size_t)k * OUT_ + gn] : 0.f;
  }
  __syncthreads();

  int wid = threadIdx.x >> 5;
  int lane = threadIdx.x & 31;
  int mt = blockIdx.x * 8 + wid;               // M-tile per wave
  int m = lane & 15;
  int h = lane >> 4;
  int klo = h * 2;

  const float* Arow = x + (size_t)(mt * 16 + m) * C_;
  const float* Brow = sT + m * C_;             // column n = nt*16 + m
  v8f acc = {0.f, 0.f, 0.f, 0.f, 0.f, 0.f, 0.f, 0.f};

  for (int k0 = 0; k0 < C_; k0 += 4) {
    v2f a, b;
    a.x = Arow[k0 + klo];
    a.y = Arow[k0 + klo + 1];
    b.x = Brow[k0 + klo];                      // ds_load_b64, no selects
    b.y = Brow[k0 + klo + 1];
    acc = wmma_f32_16x16x4(a, b, acc);
  }

  int n = nt * 16 + m;
  if (n < OUT_) {
#pragma unroll
    for (int v = 0; v < 8; ++v) {
      int row = mt * 16 + v + 8 * h;
      xt[(size_t)row * OUT_ + n] = acc[v];
    }
  }
}

// ------------------------------------------------------------- top-2 helpers
static __device__ __forceinline__ bool lessp(float a, int ia, float b, int ib) {
  return (a < b) || (a == b && ia < ib);       // jax top_k tie-break: low index
}
static __device__ __forceinline__ void top2_merge(float& d0, int& i0, float& d1,
                                                  int& i1, float e0, int j0,
                                                  float e1, int j1) {
  if (lessp(e0, j0, d0, i0)) {
    float nd1; int ni1;
    if (lessp(d0, i0, e1, j1)) { nd1 = d0; ni1 = i0; } else { nd1 = e1; ni1 = j1; }
    d0 = e0; i0 = j0; d1 = nd1; i1 = ni1;
  } else if (lessp(e0, j0, d1, i1)) {
    d1 = e0; i1 = j0;
  }
}

// ------------------------------- Gram tile via WMMA + fused distance + top-2
__global__ void nn_topk_kernel(const float* __restrict__ x,
                               const float* __restrict__ sqv,
                               int2* __restrict__ nn) {
  __shared__ float sA[16 * C_];                // 48 KB row-tile stage
  __shared__ float sSq[16];

  int bb = blockIdx.y;                         // batch
  int rowBase = blockIdx.x * 16;               // row tile within batch
  size_t gRowBase = ((size_t)bb * L_ + rowBase) * C_;

  stage_to_lds(x + gRowBase, sA, 16 * C_ / 4, threadIdx.x, blockDim.x);
  if (threadIdx.x < 16) sSq[threadIdx.x] = sqv[bb * L_ + rowBase + threadIdx.x];
  __syncthreads();

  int wid = threadIdx.x >> 5;                  // 8 waves / block
  int lane = threadIdx.x & 31;
  int m = lane & 15;
  int h = lane >> 4;
  int klo = h * 2;

  float sqA[8];
#pragma unroll
  for (int v = 0; v < 8; ++v) sqA[v] = sSq[v + 8 * h];

  float rd0[8], rd1[8];
  int ri0[8], ri1[8];
#pragma unroll
  for (int v = 0; v < 8; ++v) {
    rd0[v] = INFINITY; rd1[v] = INFINITY;
    ri0[v] = 0x7fffffff; ri1[v] = 0x7fffffff;
  }

  for (int ct = wid; ct < L_ / 16; ct += 8) {  // column tiles strided by wave
    const float* Bcol = x + ((size_t)bb * L_ + ct * 16 + m) * C_;
    float sqB = sqv[bb * L_ + ct * 16 + m];
    if (ct + 8 < L_ / 16)                      // hint next panel into cache
      __builtin_prefetch(Bcol + (size_t)128 * C_, 0, 1);

    v8f acc = {0.f, 0.f, 0.f, 0.f, 0.f, 0.f, 0.f, 0.f};
    for (int k0 = 0; k0 < C_; k0 += 4) {
      v2f a, b;
      a.x = sA[m * C_ + k0 + klo];             // ds_load from staged rows
      a.y = sA[m * C_ + k0 + klo + 1];
      b.x = Bcol[k0 + klo];
      b.y = Bcol[k0 + klo + 1];
      acc = wmma_f32_16x16x4(a, b, acc);       // G tile accumulate
    }

#pragma unroll
    for (int v = 0; v < 8; ++v) {
      float d0 = sqA[v] + sqB - 2.0f * acc[v]; // D layout: M = v + 8*half
      int i0 = ct * 16 + m;
      float d1 = INFINITY;
      int i1 = 0x7fffffff;
      // 16-lane butterfly: all lanes of a half converge to the tile top-2
      for (int msk = 1; msk < 16; msk <<= 1) {
        float e0 = __shfl_xor(d0, msk, 16);
        int   j0 = __shfl_xor(i0, msk, 16);
        float e1 = __shfl_xor(d1, msk, 16);
        int   j1 = __shfl_xor(i1, msk, 16);
        top2_merge(d0, i0, d1, i1, e0, j0, e1, j1);
      }
      top2_merge(rd0[v], ri0[v], rd1[v], ri1[v], d0, i0, d1, i1);
    }
  }

  if ((lane & 15) == 0) {                      // lane 0 of each half writes 8 rows
#pragma unroll
    for (int v = 0; v < 8; ++v) {
      int row = rowBase + v + 8 * h;
      nn[(size_t)bb * L_ + row] = make_int2(ri0[v], ri1[v]);
    }
  }
}

// ------------------- hyperedge average + scatter to node accumulator/degree
__global__ void scatter_kernel(const float* __restrict__ xt,
                               const int2* __restrict__ nn,
                               float* __restrict__ acc, int* __restrict__ deg) {
  int e = blockIdx.x;                          // hyperedge over B*L
  int b = e / L_, l = e % L_;
  int2 p = nn[e];
  int m0 = p.x, m1 = p.y, m2 = l;              // members {nn0, nn1, self}
  bool k1 = (m1 != m0);
  bool k2 = (m2 != m0) && (!k1 || m2 != m1);
  float inv = 1.0f / (float)(1 + (int)k1 + (int)k2);

  size_t base = (size_t)b * L_ * OUT_;
  int c = threadIdx.x;
  if (c < OUT_) {
    float s = xt[base + (size_t)m0 * OUT_ + c];
    if (k1) s += xt[base + (size_t)m1 * OUT_ + c];
    if (k2) s += xt[base + (size_t)m2 * OUT_ + c];
    s *= inv;                                  // edge_ft
    atomicAdd(&acc[base + (size_t)m0 * OUT_ + c], s);
    if (k1) atomicAdd(&acc[base + (size_t)m1 * OUT_ + c], s);
    if (k2) atomicAdd(&acc[base + (size_t)m2 * OUT_ + c], s);
  }
  if (threadIdx.x == 0) {
    atomicAdd(&deg[b * L_ + m0], 1);
    if (k1) atomicAdd(&deg[b * L_ + m1], 1);
    if (k2) atomicAdd(&deg[b * L_ + m2], 1);
  }
}

// ------------- node_ft = acc/deg (in place) + per-BN-channel sum / sumsq
__global__ void bnstat_kernel(float* __restrict__ acc, const int* __restrict__ deg,
                              float* __restrict__ bnSum, float* __restrict__ bnSq) {
  int o = blockIdx.x;                          // BN channel 0..158
  int b = blockIdx.y;
  size_t base = (size_t)b * L_ * OUT_ + (size_t)o * L_;  // 2048 contiguous floats
  float s = 0.f, s2 = 0.f;
  for (int t = threadIdx.x; t < L_; t += blockDim.x) {
    int n = o * L_ + t;                        // flat within (L, OUT)
    int l = n / OUT_;                          // node index for degree
    float v = acc[base + t] / (float)deg[b * L_ + l];
    acc[base + t] = v;
    s += v; s2 += v * v;
  }
  __shared__ float rs[256], rs2[256];
  rs[threadIdx.x] = s; rs2[threadIdx.x] = s2;
  __syncthreads();
  for (int st = 128; st > 0; st >>= 1) {
    if (threadIdx.x < st) {
      rs[threadIdx.x] += rs[threadIdx.x + st];
      rs2[threadIdx.x] += rs2[threadIdx.x + st];
    }
    __syncthreads();
  }
  if (threadIdx.x == 0) { atomicAdd(&bnSum[o], rs[0]); atomicAdd(&bnSq[o], rs2[0]); }
}

// ---------------------------------------------------- BN normalize + ReLU
__global__ void finalize_kernel(const float* __restrict__ acc,
                                const float* __restrict__ bnSum,
                                const float* __restrict__ bnSq,
                                const float* __restrict__ gamma,
                                const float* __restrict__ beta,
                                float* __restrict__ out) {
  size_t idx = (size_t)blockIdx.x * blockDim.x + threadIdx.x;
  const size_t per = (size_t)L_ * OUT_;
  if (idx >= (size_t)B_ * per) return;
  int o = (int)((idx % per) / L_);
  const float cnt = (float)(B_ * L_);
  float mean = bnSum[o] / cnt;
  float var = bnSq[o] / cnt - mean * mean;     // biased variance (jnp.var)
  float y = (acc[idx] - mean) * rsqrtf(var + BN_EPS);
  y = y * gamma[o] + beta[o];
  out[idx] = fmaxf(y, 0.f);
}

// --------------------------------------------------------------------------
extern "C" void kernel_launch(void* const* d_in, const int* in_sizes, int n_in,
                              void* d_out, int out_size, void* d_ws, size_t ws_size,
                              hipStream_t stream) {
  const float* x     = (const float*)d_in[0];
  const float* theta = (const float*)d_in[1];
  const float* gamma = (const float*)d_in[2];
  const float* beta  = (const float*)d_in[3];
  float* out = (float*)d_out;

  char* w = (char*)d_ws;
  auto carve = [&](size_t bytes) -> void* {
    void* p = (void*)w;
    w += (bytes + 255) & ~(size_t)255;
    return p;
  };
  const size_t featBytes = (size_t)B_ * L_ * OUT_ * sizeof(float);   // 5.2 MB
  float* xt    = (float*)carve(featBytes);
  float* sq    = (float*)carve((size_t)B_ * L_ * sizeof(float));
  int2*  nn    = (int2*) carve((size_t)B_ * L_ * sizeof(int2));
  float* acc   = (float*)carve(featBytes);
  int*   deg   = (int*)  carve((size_t)B_ * L_ * sizeof(int));
  float* bnSum = (float*)carve(2 * OUT_ * sizeof(float));
  float* bnSq  = bnSum + OUT_;

  // atomically-accumulated buffers must start at zero every call
  (void)hipMemsetAsync(acc, 0, featBytes, stream);
  (void)hipMemsetAsync(deg, 0, (size_t)B_ * L_ * sizeof(int), stream);
  (void)hipMemsetAsync(bnSum, 0, 2 * OUT_ * sizeof(float), stream);

  sq_kernel<<<(B_ * L_) / 256, 256, 0, stream>>>(x, sq);

  dim3 gGEMM(MT_ / 8, NT_);
  xt_gemm_kernel<<<gGEMM, 256, 0, stream>>>(x, theta, xt);

  dim3 gNN(L_ / 16, B_);
  nn_topk_kernel<<<gNN, 256, 0, stream>>>(x, sq, nn);

  scatter_kernel<<<B_ * L_, 160, 0, stream>>>(xt, nn, acc, deg);

  dim3 gBN(OUT_, B_);
  bnstat_kernel<<<gBN, 256, 0, stream>>>(acc, deg, bnSum, bnSq);

  size_t tot = (size_t)B_ * L_ * OUT_;
  finalize_kernel<<<(unsigned)((tot + 255) / 256), 256, 0, stream>>>(
      acc, bnSum, bnSq, gamma, beta, out);
}